// OpenCLIPer_29884382445590
// MI455X (gfx1250) — compile-verified
//
#include <hip/hip_runtime.h>
#include <hip/hip_bf16.h>

typedef float v2f __attribute__((ext_vector_type(2)));
typedef float v8f __attribute__((ext_vector_type(8)));

#define Bn 8
#define Sn 577
#define Dn 1280
#define Hn 16
#define HDn 80
#define BHn (Bn * Hn)
#define Mx (Bn * Sn)   // 4616

// ---------------------------------------------------------------------------
// Generic GEMM:  C[M,N] = A[M,K] @ B[N,K]^T (+ bias[n])
// block = 256 threads = 8 waves; tile 64(M) x 64(N), K-chunk 32.
// wave w: rows (w&3)*16, cols (w>>2)*32 (two 16x16 accumulators).
// ---------------------------------------------------------------------------
__global__ __launch_bounds__(256) void gemm_nt_bias(
    const float* __restrict__ A, int lda,
    const float* __restrict__ Bm, int ldb,
    const float* __restrict__ bias,
    float* __restrict__ C, int ldc,
    int M, int N, int K)
{
    __shared__ float As[64][33];
    __shared__ float Bs[64][33];

    const int t    = threadIdx.x;
    const int lane = t & 31;
    const int w    = t >> 5;
    const int lh   = lane >> 4;   // half-wave select
    const int lm   = lane & 15;
    const int wm   = (w & 3) * 16;           // wave row base in tile
    const int wn   = (w >> 2) * 32;          // wave col base in tile
    const int m0   = blockIdx.y * 64;
    const int n0   = blockIdx.x * 64;

    v8f c0 = {0.f,0.f,0.f,0.f,0.f,0.f,0.f,0.f};
    v8f c1 = {0.f,0.f,0.f,0.f,0.f,0.f,0.f,0.f};

    for (int k0 = 0; k0 < K; k0 += 32) {
        // load 64x32 A tile and 64x32 B tile (B stored [n][k])
        #pragma unroll
        for (int u = 0; u < 8; ++u) {
            int li  = u * 256 + t;
            int row = li >> 5, col = li & 31;
            int gm = m0 + row, gk = k0 + col;
            As[row][col] = (gm < M && gk < K) ? A[gm * lda + gk] : 0.f;
            int gn = n0 + row;
            Bs[row][col] = (gn < N && gk < K) ? Bm[gn * ldb + gk] : 0.f;
        }
        __syncthreads();

        #pragma unroll
        for (int it = 0; it < 8; ++it) {
            const int kk = it * 4;
            v2f a, b0, b1;
            a.x  = As[wm + lm][kk + 2 * lh + 0];
            a.y  = As[wm + lm][kk + 2 * lh + 1];
            b0.x = Bs[wn + lm][kk + 2 * lh + 0];
            b0.y = Bs[wn + lm][kk + 2 * lh + 1];
            b1.x = Bs[wn + 16 + lm][kk + 2 * lh + 0];
            b1.y = Bs[wn + 16 + lm][kk + 2 * lh + 1];
            c0 = __builtin_amdgcn_wmma_f32_16x16x4_f32(false, a, false, b0,
                                                       (short)0, c0, false, false);
            c1 = __builtin_amdgcn_wmma_f32_16x16x4_f32(false, a, false, b1,
                                                       (short)0, c1, false, false);
        }
        __syncthreads();
    }

    #pragma unroll
    for (int r = 0; r < 8; ++r) {
        int m = m0 + wm + r + 8 * lh;
        int n = n0 + wn + lm;
        if (m < M && n < N)
            C[m * ldc + n] = c0[r] + (bias ? bias[n] : 0.f);
        n += 16;
        if (m < M && n < N)
            C[m * ldc + n] = c1[r] + (bias ? bias[n] : 0.f);
    }
}

// ---------------------------------------------------------------------------
// attn @ v per (b,h):  AO[b, i, h*80+e] = sum_j W[bh,i,j] * V[b, j, h*80+e]
// block = 128 threads = 4 waves; tile 64(M) x 80(N), K-chunk 32.
// ---------------------------------------------------------------------------
__global__ __launch_bounds__(128) void gemm_attn_v(
    const float* __restrict__ Wt,   // [BH, S, S]
    const float* __restrict__ V,    // [B*S, D]
    float* __restrict__ AO)         // [B*S, D]
{
    __shared__ float As[64][33];
    __shared__ float Bs[32][81];

    const int t    = threadIdx.x;
    const int lane = t & 31;
    const int w    = t >> 5;        // 0..3 -> 16-row strip
    const int lh   = lane >> 4;
    const int lm   = lane & 15;
    const int m0   = blockIdx.x * 64;
    const int bh   = blockIdx.y;
    const int b    = bh / Hn;
    const int h    = bh % Hn;

    const float* Aw = Wt + (size_t)bh * Sn * Sn;
    const float* Vb = V + (size_t)b * Sn * Dn + h * HDn;
    float*       Ob = AO + (size_t)b * Sn * Dn + h * HDn;

    v8f acc[5];
    #pragma unroll
    for (int i = 0; i < 5; ++i)
        acc[i] = (v8f){0.f,0.f,0.f,0.f,0.f,0.f,0.f,0.f};

    for (int k0 = 0; k0 < Sn; k0 += 32) {
        #pragma unroll
        for (int u = 0; u < 16; ++u) {
            int li  = u * 128 + t;
            int row = li >> 5, col = li & 31;
            int gm = m0 + row, gk = k0 + col;
            As[row][col] = (gm < Sn && gk < Sn) ? Aw[gm * Sn + gk] : 0.f;
        }
        #pragma unroll
        for (int u = 0; u < 20; ++u) {
            int li  = u * 128 + t;
            int row = li / 80, col = li % 80;
            int gj = k0 + row;
            Bs[row][col] = (gj < Sn) ? Vb[gj * Dn + col] : 0.f;
        }
        __syncthreads();

        #pragma unroll
        for (int it = 0; it < 8; ++it) {
            const int kk = it * 4;
            v2f a;
            a.x = As[w * 16 + lm][kk + 2 * lh + 0];
            a.y = As[w * 16 + lm][kk + 2 * lh + 1];
            #pragma unroll
            for (int nt = 0; nt < 5; ++nt) {
                v2f bf;
                bf.x = Bs[kk + 2 * lh + 0][nt * 16 + lm];
                bf.y = Bs[kk + 2 * lh + 1][nt * 16 + lm];
                acc[nt] = __builtin_amdgcn_wmma_f32_16x16x4_f32(
                    false, a, false, bf, (short)0, acc[nt], false, false);
            }
        }
        __syncthreads();
    }

    #pragma unroll
    for (int nt = 0; nt < 5; ++nt)
        #pragma unroll
        for (int r = 0; r < 8; ++r) {
            int m = m0 + w * 16 + r + 8 * lh;
            int e = nt * 16 + lm;
            if (m < Sn) Ob[m * Dn + e] = acc[nt][r];
        }
}

// ---------------------------------------------------------------------------
// Mask pipeline: five memory-bound passes over [BH, S, S].
// m3[i,j] = 0.5*(a[i,j]/(cs[j]*rs[i]) + a[j,i]/(cs[i]*rs[j]))  (symmetric)
// ---------------------------------------------------------------------------
__global__ __launch_bounds__(256) void mask_colsum(
    const float* __restrict__ a, float* __restrict__ cs)
{
    const int bh = blockIdx.y;
    const int j  = blockIdx.x * 256 + threadIdx.x;
    if (j >= Sn) return;
    const float* m = a + (size_t)bh * Sn * Sn;
    float s = 0.f;
    for (int i = 0; i < Sn; ++i) s += m[(size_t)i * Sn + j];
    cs[bh * Sn + j] = s;
}

__device__ __forceinline__ float wave_sum(float v) {
    #pragma unroll
    for (int off = 16; off >= 1; off >>= 1) v += __shfl_xor(v, off, 32);
    return v;
}
__device__ __forceinline__ float wave_max(float v) {
    #pragma unroll
    for (int off = 16; off >= 1; off >>= 1) v = fmaxf(v, __shfl_xor(v, off, 32));
    return v;
}

__global__ __launch_bounds__(256) void mask_rowsum(
    const float* __restrict__ a, const float* __restrict__ cs,
    float* __restrict__ rs)
{
    const int bh   = blockIdx.y;
    const int lane = threadIdx.x & 31;
    const int i    = blockIdx.x * 8 + (threadIdx.x >> 5);
    if (i >= Sn) return;
    const float* m  = a + (size_t)bh * Sn * Sn + (size_t)i * Sn;
    const float* c  = cs + bh * Sn;
    float p = 0.f;
    for (int j = lane; j < Sn; j += 32) p += m[j] / c[j];
    p = wave_sum(p);
    if (lane == 0) rs[bh * Sn + i] = p;
}

__global__ __launch_bounds__(256) void mask_colmax(
    const float* __restrict__ a, const float* __restrict__ cs,
    const float* __restrict__ rs, float* __restrict__ cm)
{
    const int bh   = blockIdx.y;
    const int lane = threadIdx.x & 31;
    const int j    = blockIdx.x * 8 + (threadIdx.x >> 5);
    if (j >= Sn) return;
    const float* m = a + (size_t)bh * Sn * Sn;
    const float* c = cs + bh * Sn;
    const float* r = rs + bh * Sn;
    const float cj = c[j], rj = r[j];
    float v = -3.4e38f;
    for (int k = lane; k < Sn; k += 32) {
        float m3 = 0.5f * (m[(size_t)j * Sn + k] / (c[k] * rj) +
                           m[(size_t)k * Sn + j] / (cj * r[k]));
        v = fmaxf(v, m3);
    }
    v = wave_max(v);
    if (lane == 0) cm[bh * Sn + j] = v;
}

__global__ __launch_bounds__(256) void mask_rowsum2(
    const float* __restrict__ a, const float* __restrict__ cs,
    const float* __restrict__ rs, const float* __restrict__ cm,
    float* __restrict__ rs2)
{
    const int bh   = blockIdx.y;
    const int lane = threadIdx.x & 31;
    const int i    = blockIdx.x * 8 + (threadIdx.x >> 5);
    if (i >= Sn) return;
    const float* m = a + (size_t)bh * Sn * Sn;
    const float* c = cs + bh * Sn;
    const float* r = rs + bh * Sn;
    const float* x = cm + bh * Sn;
    const float ci = c[i], ri = r[i];
    float p = 0.f;
    for (int j = lane; j < Sn; j += 32) {
        float m3 = 0.5f * (m[(size_t)i * Sn + j] / (c[j] * ri) +
                           m[(size_t)j * Sn + i] / (ci * r[j]));
        p += fmaxf(m3 - 0.2f * x[j], 0.f);
    }
    p = wave_sum(p);
    if (lane == 0) rs2[bh * Sn + i] = p;
}

__global__ __launch_bounds__(256) void mask_write(
    const float* __restrict__ a, const float* __restrict__ cs,
    const float* __restrict__ rs, const float* __restrict__ cm,
    const float* __restrict__ rs2, float* __restrict__ out)
{
    const int bh = blockIdx.z;
    const int i  = blockIdx.y;
    const int j  = blockIdx.x * 256 + threadIdx.x;
    if (j >= Sn) return;
    const float* m = a + (size_t)bh * Sn * Sn;
    const float* c = cs + bh * Sn;
    const float* r = rs + bh * Sn;
    float m3 = 0.5f * (m[(size_t)i * Sn + j] / (c[j] * r[i]) +
                       m[(size_t)j * Sn + i] / (c[i] * r[j]));
    float w = fmaxf(m3 - 0.2f * cm[bh * Sn + j], 0.f) / rs2[bh * Sn + i];
    out[(size_t)bh * Sn * Sn + (size_t)i * Sn + j] = w;
}

// ---------------------------------------------------------------------------
extern "C" void kernel_launch(void* const* d_in, const int* in_sizes, int n_in,
                              void* d_out, int out_size, void* d_ws, size_t ws_size,
                              hipStream_t stream) {
    const float* x      = (const float*)d_in[0];
    const float* mask   = (const float*)d_in[1];
    const float* w_in   = (const float*)d_in[2];
    const float* b_in   = (const float*)d_in[3];
    const float* w_out  = (const float*)d_in[4];
    const float* b_out  = (const float*)d_in[5];

    float* out_attn = (float*)d_out;                              // [B,S,D]
    float* out_wts  = out_attn + (size_t)Bn * Sn * Dn;            // [BH,S,S]

    float* ws  = (float*)d_ws;
    float* V   = ws;                                   // [B*S, D]
    float* AO  = V + (size_t)Mx * Dn;                  // [B*S, D]
    float* cs  = AO + (size_t)Mx * Dn;                 // [BH*S]
    float* rs  = cs + BHn * Sn;
    float* cm  = rs + BHn * Sn;
    float* rs2 = cm + BHn * Sn;

    // 1) V projection: V = x @ Wv^T + bv   (only the V third of in_proj used)
    gemm_nt_bias<<<dim3(Dn / 64, (Mx + 63) / 64), 256, 0, stream>>>(
        x, Dn, w_in + (size_t)2 * Dn * Dn, Dn, b_in + 2 * Dn, V, Dn, Mx, Dn, Dn);

    // 2) mask pipeline -> attn_weights (written straight into d_out slice)
    mask_colsum<<<dim3((Sn + 255) / 256, BHn), 256, 0, stream>>>(mask, cs);
    mask_rowsum<<<dim3((Sn + 7) / 8, BHn), 256, 0, stream>>>(mask, cs, rs);
    mask_colmax<<<dim3((Sn + 7) / 8, BHn), 256, 0, stream>>>(mask, cs, rs, cm);
    mask_rowsum2<<<dim3((Sn + 7) / 8, BHn), 256, 0, stream>>>(mask, cs, rs, cm, rs2);
    mask_write<<<dim3((Sn + 255) / 256, Sn, BHn), 256, 0, stream>>>(
        mask, cs, rs, cm, rs2, out_wts);

    // 3) attn_weights @ V  -> AO in [B,S,H*HD] layout (no repack pass needed)
    gemm_attn_v<<<dim3((Sn + 63) / 64, BHn), 128, 0, stream>>>(out_wts, V, AO);

    // 4) output projection
    gemm_nt_bias<<<dim3(Dn / 64, (Mx + 63) / 64), 256, 0, stream>>>(
        AO, Dn, w_out, Dn, b_out, out_attn, Dn, Mx, Dn, Dn);
}